// ODELinear_73547019977229
// MI455X (gfx1250) — compile-verified
//
#include <hip/hip_runtime.h>
#include <cstddef>

typedef __attribute__((ext_vector_type(2))) float v2f;
typedef __attribute__((ext_vector_type(8))) float v8f;

#define Bdim 8
#define Ldim 96
#define Hdim 8
#define Edim 64
#define EHdim 128

// Half of the B*H=64 rows per workgroup (2 WGs per trajectory index)
#define MROWS 32

// LDS leading dims padded +4 floats so strided (column) fragment loads
// advance 4 banks per row -> 16 lanes hit 16 distinct banks.
#define LDY 68    // 64-wide buffers (state, k)
#define LDH 132   // 128-wide buffer (hidden)

// fast tanh: 1 - 2/(exp(2x)+1)  (v_exp_f32 + v_rcp_f32; saturates to +-1)
__device__ __forceinline__ float fast_tanh(float v) {
  const float e = __expf(v + v);
  return 1.0f - 2.0f * __builtin_amdgcn_rcpf(e + 1.0f);
}

__launch_bounds__(256, 1)
__global__ void ode_bidir_wmma(const float* __restrict__ x,
                               const float* __restrict__ tsg,
                               const float* __restrict__ W1g,
                               const float* __restrict__ b1g,
                               const float* __restrict__ W2g,
                               const float* __restrict__ b2g,
                               float* __restrict__ out)
{
  __shared__ float sYin[MROWS * LDY];   // f() input
  __shared__ float sH[MROWS * LDH];     // hidden activations
  __shared__ float sKt[MROWS * LDY];    // f() output
  __shared__ float sTs[Ldim];

  const int tid  = threadIdx.x;
  const int idx  = blockIdx.x;          // trajectory start index, 0..95
  const int half = blockIdx.y;          // which half of the 64 rows
  const int w    = tid >> 5;            // wave id 0..7
  const int lane = tid & 31;
  const int hiL  = lane >> 4;           // half-wave (WMMA layout)
  const int ln   = lane & 15;

  if (tid < Ldim) sTs[tid] = tsg[tid];

  // ---- weight B-fragments: loop-invariant, held in VGPRs for whole kernel ----
  // matmul1: wave w owns output col-slab n0 = w*16 (of 128)
  // matmul2: wave w owns tile (mt2 = w>>2, nt2 = w&3) of the 32x64 output
  const int n0  = w << 4;
  const int mt2 = w >> 2;
  const int nt2 = (w & 3) << 4;

  v2f bW1[16];  // K = 64  -> 16 frags
  v2f bW2[32];  // K = 128 -> 32 frags
  #pragma unroll
  for (int kk = 0; kk < 16; ++kk) {
    bW1[kk].x = W1g[(4 * kk + 2 * hiL)     * EHdim + n0 + ln];
    bW1[kk].y = W1g[(4 * kk + 2 * hiL + 1) * EHdim + n0 + ln];
  }
  #pragma unroll
  for (int kk = 0; kk < 32; ++kk) {
    bW2[kk].x = W2g[(4 * kk + 2 * hiL)     * Edim + nt2 + ln];
    bW2[kk].y = W2g[(4 * kk + 2 * hiL + 1) * Edim + nt2 + ln];
  }
  const float bias1 = b1g[n0 + ln];   // per-lane column bias, C-init for WMMA
  const float bias2 = b2g[nt2 + ln];

  // ---- per-thread element ownership: 8 threads per row, 8 elems each ----
  const int rloc    = tid >> 3;                 // 0..31 local row
  const int rowg    = half * MROWS + rloc;      // 0..63 = b*H + h
  const int bb      = rowg >> 3;
  const int hh      = rowg & 7;
  const int colbase = (tid & 7) * 8;

  float x0v[8], y[8], k1[8], k2[8], accm[8];
  {
    const size_t xbase = (((size_t)bb * Ldim + idx) * Hdim + hh) * Edim + colbase;
    #pragma unroll
    for (int i = 0; i < 8; ++i) x0v[i] = x[xbase + i];
  }
  __syncthreads();

  // ---- f(): sYin -> sKt through WMMA f32 16x16x4, weights in registers ----
  auto feval = [&]() {
    __syncthreads();                       // sYin writes visible
    // matmul1: (32 x 64) @ (64 x 128), bias in C, tanh on store
    #pragma unroll
    for (int mt = 0; mt < 2; ++mt) {
      v8f acc;
      #pragma unroll
      for (int r = 0; r < 8; ++r) acc[r] = bias1;
      #pragma unroll
      for (int kk = 0; kk < 16; ++kk) {
        v2f a = *(const v2f*)&sYin[(mt * 16 + ln) * LDY + 4 * kk + 2 * hiL];
        acc = __builtin_amdgcn_wmma_f32_16x16x4_f32(false, a, false, bW1[kk],
                                                    (short)0, acc, false, false);
      }
      #pragma unroll
      for (int r = 0; r < 8; ++r)
        sH[(mt * 16 + r + 8 * hiL) * LDH + n0 + ln] = fast_tanh(acc[r]);
    }
    __syncthreads();
    // matmul2: (32 x 128) @ (128 x 64), bias in C
    {
      v8f acc;
      #pragma unroll
      for (int r = 0; r < 8; ++r) acc[r] = bias2;
      #pragma unroll
      for (int kk = 0; kk < 32; ++kk) {
        v2f a = *(const v2f*)&sH[(mt2 * 16 + ln) * LDH + 4 * kk + 2 * hiL];
        acc = __builtin_amdgcn_wmma_f32_16x16x4_f32(false, a, false, bW2[kk],
                                                    (short)0, acc, false, false);
      }
      #pragma unroll
      for (int r = 0; r < 8; ++r)
        sKt[(mt2 * 16 + r + 8 * hiL) * LDY + nt2 + ln] = acc[r];
    }
    __syncthreads();                       // sKt visible
  };

  // ---- one 3/8-rule RK4 step, dt uniform per block ----
  auto rk4 = [&](float dt) {
    const float dt3 = dt * (1.0f / 3.0f);
    #pragma unroll
    for (int i = 0; i < 8; ++i) sYin[rloc * LDY + colbase + i] = y[i];
    feval();                               // k1
    #pragma unroll
    for (int i = 0; i < 8; ++i) {
      const float kt = sKt[rloc * LDY + colbase + i];
      k1[i] = kt; accm[i] = kt;
      sYin[rloc * LDY + colbase + i] = y[i] + dt3 * kt;
    }
    feval();                               // k2
    #pragma unroll
    for (int i = 0; i < 8; ++i) {
      const float kt = sKt[rloc * LDY + colbase + i];
      k2[i] = kt; accm[i] += 3.0f * kt;
      sYin[rloc * LDY + colbase + i] = y[i] + dt * (kt - (1.0f / 3.0f) * k1[i]);
    }
    feval();                               // k3
    #pragma unroll
    for (int i = 0; i < 8; ++i) {
      const float kt = sKt[rloc * LDY + colbase + i];
      accm[i] += 3.0f * kt;
      sYin[rloc * LDY + colbase + i] = y[i] + dt * (k1[i] - k2[i] + kt);
    }
    feval();                               // k4
    #pragma unroll
    for (int i = 0; i < 8; ++i) {
      const float kt = sKt[rloc * LDY + colbase + i];
      accm[i] += kt;
      y[i] += dt * 0.125f * accm[i];
    }
  };

  // ---- emit state at time index j: out[b, idx, j, h, e, {x0, x1}] ----
  auto emit = [&](int j) {
    float2* o2 = (float2*)out;
    const size_t base =
        ((((size_t)bb * Ldim + idx) * Ldim + j) * Hdim + hh) * Edim + colbase;
    #pragma unroll
    for (int i = 0; i < 8; ++i) {
      float2 v; v.x = x0v[i]; v.y = y[i];
      o2[base + i] = v;
    }
  };

  // ---- forward: t_idx -> t_95 ----
  #pragma unroll
  for (int i = 0; i < 8; ++i) y[i] = x0v[i];
  emit(idx);
  for (int s = idx; s < Ldim - 1; ++s) {
    rk4(sTs[s + 1] - sTs[s]);
    emit(s + 1);
  }

  // ---- backward: t_idx -> t_0 ----
  #pragma unroll
  for (int i = 0; i < 8; ++i) y[i] = x0v[i];
  for (int s = idx; s >= 1; --s) {
    rk4(sTs[s - 1] - sTs[s]);              // negative dt
    emit(s - 1);
  }
}

extern "C" void kernel_launch(void* const* d_in, const int* in_sizes, int n_in,
                              void* d_out, int out_size, void* d_ws, size_t ws_size,
                              hipStream_t stream) {
  const float* x  = (const float*)d_in[0];
  const float* ts = (const float*)d_in[1];
  const float* W1 = (const float*)d_in[2];
  const float* b1 = (const float*)d_in[3];
  const float* W2 = (const float*)d_in[4];
  const float* b2 = (const float*)d_in[5];
  float* out = (float*)d_out;

  dim3 grid(Ldim, 2);   // 96 trajectory indices x 2 row-halves
  dim3 block(256);      // 8 waves (wave32)
  ode_bidir_wmma<<<grid, block, 0, stream>>>(x, ts, W1, b1, W2, b2, out);
}